// snn_mnist_29755533427594
// MI455X (gfx1250) — compile-verified
//
#include <hip/hip_runtime.h>

// ---------------------------------------------------------------------------
// SNN + STDP on MI455X (gfx1250, wave32, WMMA).
// Exact-precision path: V_WMMA_F32_16X16X4_F32 for all three GEMMs per step.
// Per step: [xpose+pretrace] -> [forward LIF GEMM] -> [STDP GEMM + clip].
// Serial over T=100 (W(t) feeds GEMM(t+1)): 3 kernels/step on one stream.
// ---------------------------------------------------------------------------

typedef __attribute__((ext_vector_type(2))) float v2f;
typedef __attribute__((ext_vector_type(8))) float v8f;

#define T_STEPS   100
#define BATCH     256
#define IN_SZ     784
#define NN        512

#define ALPHA     0.9f
#define BETA      0.8f
#define BETA_P    0.9f
#define BETA_M    0.9f
#define THR       1.0f
#define A_PLUS    0.008f
#define A_MINUS   0.005f

// workspace layout (float offsets)
#define OFF_SYN   0
#define OFF_MEM   (BATCH*NN)
#define OFF_SPKT  (2*BATCH*NN)
#define OFF_POSTT (3*BATCH*NN)
#define OFF_PRET  (4*BATCH*NN)
#define OFF_XT    (4*BATCH*NN + IN_SZ*BATCH)
#define WS_FLOATS (4*BATCH*NN + 2*IN_SZ*BATCH)   // 925,696 floats (~3.6 MB)

// fp32 WMMA wrapper: D(16x16) += A(16x4) * B(4x16), all f32, wave32.
__device__ __forceinline__ v8f wmma_f32(v2f a, v2f b, v8f c) {
    return __builtin_amdgcn_wmma_f32_16x16x4_f32(
        /*neg_a=*/false, a, /*neg_b=*/false, b,
        /*c_mod=*/(short)0, c, /*reuse_a=*/false, /*reuse_b=*/false);
}

// ---- init: zero recurrent state, copy W_in into working W (= d_out) --------
__global__ void __launch_bounds__(256)
snn_init(float* __restrict__ ws, float* __restrict__ W,
         const float* __restrict__ W_in) {
    int i = blockIdx.x * 256 + threadIdx.x;
    if (i < WS_FLOATS) ws[i] = 0.0f;
    if (i < NN * IN_SZ) W[i] = W_in[i];
}

// ---- per step: xT[i][b] = x_t[b][i]; preT = BETA_P*preT + x ----------------
// Also prefetch next step's image slab into L2 (global_prefetch_b8).
__global__ void __launch_bounds__(256)
snn_xpose_pretrace(const float* __restrict__ x_t,
                   const float* __restrict__ x_next,   // nullptr on last step
                   float* __restrict__ xT,
                   float* __restrict__ preT) {
    int i = blockIdx.x * 256 + threadIdx.x;        // flat over IN_SZ*BATCH
    int in_idx = i >> 8;                            // / BATCH
    int b      = i & (BATCH - 1);                   // % BATCH
    float v = x_t[b * IN_SZ + in_idx];
    xT[i]   = v;
    preT[i] = BETA_P * preT[i] + v;
    if (x_next) __builtin_prefetch(x_next + i, 0, 1);
}

// ---- forward: I = x_t * W^T (tiles 16x16 over BxN), LIF update, traces -----
// One wave32 per 16x16 output tile; K=IN_SZ in steps of 4 (fp32 WMMA).
__global__ void __launch_bounds__(256)
snn_forward(const float* __restrict__ x_t, const float* __restrict__ W,
            float* __restrict__ syn, float* __restrict__ mem,
            float* __restrict__ spkT, float* __restrict__ postT) {
    const int lane = threadIdx.x & 31;
    const int wid  = threadIdx.x >> 5;
    const int g    = blockIdx.x * 8 + wid;         // 512 waves = 16 x 32 tiles
    const int b0   = (g >> 5) << 4;                // batch tile base
    const int n0   = (g & 31) << 4;                // neuron tile base
    const int m    = lane & 15;                    // A row / B col within tile
    const int kp   = (lane >> 4) << 1;             // K sub-pair: 0 or 2

    // A[m][k] = x_t[b0+m][k]      (K contiguous -> float2 loads)
    // B[k][n] = W^T[k][n] = W[n0+n][k] (K contiguous -> float2 loads)
    const float* arow = x_t + (size_t)(b0 + m) * IN_SZ + kp;
    const float* brow = W   + (size_t)(n0 + m) * IN_SZ + kp;

    v8f acc = {};
#pragma unroll 4
    for (int k = 0; k < IN_SZ; k += 4) {
        v2f a = *(const v2f*)(arow + k);
        v2f b = *(const v2f*)(brow + k);
        acc = wmma_f32(a, b, acc);
    }

    // C/D layout: lane -> col n0+(lane&15); vgpr v -> row b0 + v + 8*(lane>>4)
    const int col = n0 + (lane & 15);
    const int rb  = b0 + ((lane >> 4) << 3);
#pragma unroll
    for (int v = 0; v < 8; ++v) {
        int b   = rb + v;
        int idx = b * NN + col;
        float s  = syn[idx];
        float mm = mem[idx];
        float reset = (mm > THR) ? 1.0f : 0.0f;
        s  = ALPHA * s + acc[v];
        mm = BETA * mm + s - reset * THR;
        float spk = (mm > THR) ? 1.0f : 0.0f;
        syn[idx] = s;
        mem[idx] = mm;
        int tidx = col * BATCH + b;                // transposed state for STDP
        spkT[tidx]  = spk;
        postT[tidx] = BETA_M * postT[tidx] + spk;
    }
}

// ---- STDP: W = clip(W + A+ * spk^T*preT - A- * postT^T*x, 0, 1) ------------
// Tiles 16x16 over N x IN; K = BATCH. Two fp32-WMMA GEMMs per tile.
__global__ void __launch_bounds__(256)
snn_stdp(float* __restrict__ W,
         const float* __restrict__ spkT, const float* __restrict__ preT,
         const float* __restrict__ postT, const float* __restrict__ xT) {
    const int lane = threadIdx.x & 31;
    const int wid  = threadIdx.x >> 5;
    const int g    = blockIdx.x * 8 + wid;         // 1568 waves = 32 x 49 tiles
    const int n0   = (g / (IN_SZ / 16)) << 4;
    const int i0   = (g % (IN_SZ / 16)) << 4;
    const int m    = lane & 15;
    const int kp   = (lane >> 4) << 1;

    // All four operands stored cols-major x BATCH -> K(=b) contiguous.
    const float* a1 = spkT  + (size_t)(n0 + m) * BATCH + kp;  // A1[n][b]
    const float* a2 = postT + (size_t)(n0 + m) * BATCH + kp;  // A2[n][b]
    const float* b1 = preT  + (size_t)(i0 + m) * BATCH + kp;  // B1[b][i]
    const float* b2 = xT    + (size_t)(i0 + m) * BATCH + kp;  // B2[b][i]

    v8f acc1 = {}, acc2 = {};
#pragma unroll 4
    for (int k = 0; k < BATCH; k += 4) {
        v2f av1 = *(const v2f*)(a1 + k);
        v2f bv1 = *(const v2f*)(b1 + k);
        acc1 = wmma_f32(av1, bv1, acc1);
        v2f av2 = *(const v2f*)(a2 + k);
        v2f bv2 = *(const v2f*)(b2 + k);
        acc2 = wmma_f32(av2, bv2, acc2);
    }

    const int col = i0 + (lane & 15);
    const int rn  = n0 + ((lane >> 4) << 3);
#pragma unroll
    for (int v = 0; v < 8; ++v) {
        int n = rn + v;
        size_t idx = (size_t)n * IN_SZ + col;
        float w = W[idx] + A_PLUS * acc1[v] - A_MINUS * acc2[v];
        w = fminf(fmaxf(w, 0.0f), 1.0f);
        W[idx] = w;
    }
}

extern "C" void kernel_launch(void* const* d_in, const int* in_sizes, int n_in,
                              void* d_out, int out_size, void* d_ws, size_t ws_size,
                              hipStream_t stream) {
    const float* image = (const float*)d_in[0];   // (T, B, IN) fp32 {0,1}
    const float* W_in  = (const float*)d_in[1];   // (N, IN) fp32
    float* W  = (float*)d_out;                    // working + final W (N, IN)
    float* ws = (float*)d_ws;

    float* syn   = ws + OFF_SYN;
    float* mem   = ws + OFF_MEM;
    float* spkT  = ws + OFF_SPKT;
    float* postT = ws + OFF_POSTT;
    float* preT  = ws + OFF_PRET;
    float* xT    = ws + OFF_XT;

    snn_init<<<(WS_FLOATS + 255) / 256, 256, 0, stream>>>(ws, W, W_in);

    const size_t step_elems = (size_t)BATCH * IN_SZ;
    for (int t = 0; t < T_STEPS; ++t) {
        const float* x_t    = image + (size_t)t * step_elems;
        const float* x_next = (t + 1 < T_STEPS) ? (x_t + step_elems) : nullptr;

        snn_xpose_pretrace<<<(IN_SZ * BATCH) / 256, 256, 0, stream>>>(
            x_t, x_next, xT, preT);

        snn_forward<<<(NN / 16) * (BATCH / 16) / 8, 256, 0, stream>>>(
            x_t, W, syn, mem, spkT, postT);

        snn_stdp<<<(NN / 16) * (IN_SZ / 16) / 8, 256, 0, stream>>>(
            W, spkT, preT, postT, xT);
    }
}